// JobSchedulerGNN_81363860456051
// MI455X (gfx1250) — compile-verified
//
#include <hip/hip_runtime.h>
#include <hip/hip_bf16.h>

// ---------------------------------------------------------------------------
// JobSchedulerGNN on MI455X (gfx1250, wave32)
//   layer: agg = segment_sum(h[src], dst);  h' = relu(agg@W_rel + h@W_root + b)
//   heads: alloc = h2@Wa+ba  [N,2];  order = h2@Wo+bo  [N,1]
// All fp32. GEMMs use V_WMMA_F32_16X16X4_F32; scatter uses global fp32 atomics.
// ---------------------------------------------------------------------------

#define NN      50000
#define EE      600000
#define DH      128
#define NTILES  (NN / 16)        // 3125 exact

typedef float v2f __attribute__((ext_vector_type(2)));
typedef float v4f __attribute__((ext_vector_type(4)));
typedef float v8f __attribute__((ext_vector_type(8)));

// ---------------------------------------------------------------- zero fill
__global__ void gnn_zero_f32(float* __restrict__ p, int n4) {
    int i = blockIdx.x * blockDim.x + threadIdx.x;
    if (i < n4) {
        v4f z = {0.f, 0.f, 0.f, 0.f};
        ((v4f*)p)[i] = z;
    }
}

// ------------------------------------------------------------- scatter-add
// agg[dst] += h[src]; 32 threads per edge, 4 floats per thread.
__global__ void gnn_scatter_add(const float* __restrict__ h,
                                const int* __restrict__ src,
                                const int* __restrict__ dst,
                                float* __restrict__ agg) {
    int gid = blockIdx.x * blockDim.x + threadIdx.x;
    int e = gid >> 5;
    if (e >= EE) return;
    int c = (gid & 31) << 2;
    int s = src[e];
    int d = dst[e];
    v4f v = *(const v4f*)(h + (long)s * DH + c);
    float* out = agg + (long)d * DH + c;
    unsafeAtomicAdd(out + 0, v.x);
    unsafeAtomicAdd(out + 1, v.y);
    unsafeAtomicAdd(out + 2, v.z);
    unsafeAtomicAdd(out + 3, v.w);
}

// ----------------------------------------------------- fused WMMA GEMM layer
// out = relu(Arel @ Wrel + Aroot @ Wroot + bias)
//   Arel/Aroot: [NN,128] row-major, Wrel/Wroot: [128,128] row-major.
// Block: 256 thr (8 waves). blockIdx.x -> group of 8 M-tiles (16 rows each),
// blockIdx.y -> which 64-column half of the output (keeps LDS at 64 KB).
// W^T halves staged in LDS with XOR swizzle (k-pair granularity) so staging
// stores and B-fragment ds_load_b64s are bank-conflict free.
__global__ __launch_bounds__(256)
void gnn_gemm_bias_relu(const float* __restrict__ Arel,
                        const float* __restrict__ Aroot,
                        const float* __restrict__ Wrel,
                        const float* __restrict__ Wroot,
                        const float* __restrict__ bias,
                        float* __restrict__ out) {
    __shared__ float lrel[64 * 128];    // W_rel^T  (cols hn*64..hn*64+63)
    __shared__ float lroot[64 * 128];   // W_root^T

    const int hn = blockIdx.y;          // column half: 0 or 1
    const int t  = threadIdx.x;

    // ---- stage transposed weight halves into LDS (coalesced global reads)
    for (int idx = t; idx < 64 * 128; idx += 256) {
        int k  = idx >> 6;              // 0..127
        int nn = idx & 63;              // 0..63  (local output column)
        int sw = nn * 128 + (((((k >> 1) ^ nn) & 63) << 1) | (k & 1));
        lrel[sw]  = Wrel [k * DH + hn * 64 + nn];
        lroot[sw] = Wroot[k * DH + hn * 64 + nn];
    }
    __syncthreads();

    const int wave = t >> 5;
    const int lane = t & 31;
    const int m16  = lane & 15;             // A: row M   | B: col N
    const int koff = (lane >> 4) << 1;      // K offset 0 / 2 (ISA frag layout)

    int tile = blockIdx.x * 8 + wave;
    if (tile >= NTILES) return;

    const float* ar = Arel  + ((long)tile * 16 + m16) * DH;
    const float* ax = Aroot + ((long)tile * 16 + m16) * DH;

    v8f c[4];
    v8f z = {0.f, 0.f, 0.f, 0.f, 0.f, 0.f, 0.f, 0.f};
#pragma unroll
    for (int nt = 0; nt < 4; ++nt) c[nt] = z;

    // K = 128 -> 32 steps of k=4; 4 local n-tiles; 2 GEMMs share accumulators
#pragma unroll 4
    for (int ks = 0; ks < 32; ++ks) {
        int kk = ks * 4 + koff;                       // even
        v2f a_rel = *(const v2f*)(ar + kk);
        v2f a_rt  = *(const v2f*)(ax + kk);
#pragma unroll
        for (int nt = 0; nt < 4; ++nt) {
            int nn = nt * 16 + m16;
            int sw = nn * 128 + ((((kk >> 1) ^ nn) & 63) << 1);
            v2f b_rel = *(const v2f*)(lrel + sw);
            v2f b_rt  = *(const v2f*)(lroot + sw);
            c[nt] = __builtin_amdgcn_wmma_f32_16x16x4_f32(
                        false, a_rel, false, b_rel, (short)0, c[nt], false, false);
            c[nt] = __builtin_amdgcn_wmma_f32_16x16x4_f32(
                        false, a_rt,  false, b_rt,  (short)0, c[nt], false, false);
        }
    }

    // ---- bias + relu + store (C layout: M = r + 8*(lane>=16), N = lane&15)
    const int mbase = tile * 16 + ((lane >> 4) << 3);
#pragma unroll
    for (int nt = 0; nt < 4; ++nt) {
        int n = hn * 64 + nt * 16 + m16;
        float bv = bias[n];
#pragma unroll
        for (int r = 0; r < 8; ++r) {
            float v = c[nt][r] + bv;
            out[(long)(mbase + r) * DH + n] = v > 0.f ? v : 0.f;
        }
    }
}

// ------------------------------------------------------------- output heads
// alloc[row,0:2] = h2[row]@Wa + ba ; order[row] = h2[row]@Wo + bo
// 2 lanes per row, 64-length partial dots, shfl_xor combine.
__global__ void gnn_heads(const float* __restrict__ h2,
                          const float* __restrict__ Wa,
                          const float* __restrict__ ba,
                          const float* __restrict__ Wo,
                          const float* __restrict__ bo,
                          float* __restrict__ out) {
    int t = blockIdx.x * blockDim.x + threadIdx.x;
    int row  = t >> 1;
    int half = t & 1;
    if (row >= NN) return;

    const float* h = h2 + (long)row * DH + half * 64;
    float a0 = 0.f, a1 = 0.f, oo = 0.f;
#pragma unroll 8
    for (int k = 0; k < 64; ++k) {
        float v = h[k];
        int kg = half * 64 + k;
        a0 += v * Wa[kg * 2 + 0];
        a1 += v * Wa[kg * 2 + 1];
        oo += v * Wo[kg];
    }
    a0 += __shfl_xor(a0, 1);
    a1 += __shfl_xor(a1, 1);
    oo += __shfl_xor(oo, 1);
    if (half == 0) {
        out[row * 2 + 0]  = a0 + ba[0];
        out[row * 2 + 1]  = a1 + ba[1];
        out[2 * NN + row] = oo + bo[0];
    }
}

// ---------------------------------------------------------------------------
extern "C" void kernel_launch(void* const* d_in, const int* in_sizes, int n_in,
                              void* d_out, int out_size, void* d_ws, size_t ws_size,
                              hipStream_t stream) {
    const float* x      = (const float*)d_in[0];
    const int*   ei     = (const int*)  d_in[1];   // [2,E]: row0=src, row1=dst
    const float* W1rel  = (const float*)d_in[2];
    const float* b1     = (const float*)d_in[3];
    const float* W1root = (const float*)d_in[4];
    const float* W2rel  = (const float*)d_in[5];
    const float* b2     = (const float*)d_in[6];
    const float* W2root = (const float*)d_in[7];
    const float* Wa     = (const float*)d_in[8];
    const float* ba     = (const float*)d_in[9];
    const float* Wo     = (const float*)d_in[10];
    const float* bo     = (const float*)d_in[11];
    float* out = (float*)d_out;

    float* ws  = (float*)d_ws;
    float* agg = ws;                       // [NN*DH], reused for both layers
    float* h1  = ws + (size_t)NN * DH;     // [NN*DH]
    float* h2  = ws + (size_t)2 * NN * DH; // [NN*DH]
    const int* src = ei;
    const int* dst = ei + EE;

    const int n4 = NN * DH / 4;
    dim3 gemm_grid((NTILES + 7) / 8, 2);

    // ---- layer 1
    gnn_zero_f32<<<(n4 + 255) / 256, 256, 0, stream>>>(agg, n4);
    gnn_scatter_add<<<(EE * 32) / 256, 256, 0, stream>>>(x, src, dst, agg);
    gnn_gemm_bias_relu<<<gemm_grid, 256, 0, stream>>>(agg, x, W1rel, W1root, b1, h1);

    // ---- layer 2
    gnn_zero_f32<<<(n4 + 255) / 256, 256, 0, stream>>>(agg, n4);
    gnn_scatter_add<<<(EE * 32) / 256, 256, 0, stream>>>(h1, src, dst, agg);
    gnn_gemm_bias_relu<<<gemm_grid, 256, 0, stream>>>(agg, h1, W2rel, W2root, b2, h2);

    // ---- heads
    gnn_heads<<<(2 * NN + 255) / 256, 256, 0, stream>>>(h2, Wa, ba, Wo, bo, out);
}